// ADSTFT_45956150067753
// MI455X (gfx1250) — compile-verified
//
#include <hip/hip_runtime.h>
#include <math.h>

// Problem constants (from the reference)
#define B_    4
#define L_    80000
#define N_    512
#define F_    257
#define T_    311
#define M_    (B_ * T_)      // 1244 rows (b,t)
#define MPAD  1248           // 78 * 16
#define JN    514            // 2*F interleaved re/im columns
#define JPAD  528            // 33 * 16
#define KDIM  512
#define MT    (MPAD / 16)    // 78 row tiles
#define NT    (JPAD / 16)    // 33 col tiles

#define TWO_PI 6.283185307179586f
#define EPSF   1.1920928955078125e-7f   // finfo(float32).eps

typedef float v2f __attribute__((ext_vector_type(2)));
typedef float v8f __attribute__((ext_vector_type(8)));

// Stage 1: G[m][n] = folded(b,t,n) * tap(n,t)   (padded rows -> 0)
__global__ void build_g(const float* __restrict__ x,
                        const float* __restrict__ win,
                        const float* __restrict__ strd,
                        float* __restrict__ g) {
  int e = blockIdx.x * blockDim.x + threadIdx.x;
  if (e >= MPAD * KDIM) return;
  int m = e >> 9;          // / 512
  int n = e & 511;
  float out = 0.0f;
  if (m < M_) {
    float wl = fminf(fmaxf(win[0], (float)N_ / 20.0f), (float)N_);
    float st = fminf(fmaxf(strd[0], 0.0f), (float)N_);
    int b = m / T_;
    int t = m - b * T_;
    float frame = st * (float)t;           // frames[t] = st * t
    float fl = floorf(frame);
    float frac = frame - fl;
    int idx = (int)fl + n;
    float xv = 0.0f;
    if (idx >= 0 && idx < L_) xv = x[(size_t)b * L_ + idx];
    float base = (float)n - frac;
    float hi = ceilf(((float)(N_ - 1) + wl) * 0.5f);
    float lo = floorf(((float)(N_ - 1) - wl) * 0.5f);
    float tap = 0.5f - 0.5f * cosf(TWO_PI * (base + (wl - (float)(N_ - 1)) * 0.5f) / wl);
    if (base >= hi || base <= lo) tap = 0.0f;
    tap *= 2.0f / (float)N_;
    out = xv * tap;
  }
  g[e] = out;
}

// Stage 2: W^T[j][k]: j even -> cos(2*pi*f*k/N), j odd -> -sin(...), f = j/2.
// Stored K-contiguous so the WMMA B operand loads are b64 along K.
__global__ void build_w(float* __restrict__ w) {
  int e = blockIdx.x * blockDim.x + threadIdx.x;
  if (e >= JPAD * KDIM) return;
  int j = e >> 9;
  int n = e & 511;
  float val = 0.0f;
  if (j < JN) {
    int f = j >> 1;
    int p = (f * n) & (N_ - 1);            // exact phase reduction mod 512
    float th = (float)p * (TWO_PI / (float)N_);
    val = (j & 1) ? -sinf(th) : cosf(th);
  }
  w[e] = val;
}

// Stage 3: one wave per 16x16 output tile; 128 x v_wmma_f32_16x16x4_f32.
// Epilogue rotates by shift[t,f] and emits spec + complex stft.
__global__ void __launch_bounds__(32)
wmma_dft(const float* __restrict__ g, const float* __restrict__ w,
         const float* __restrict__ strd, float* __restrict__ out) {
  const int lane = threadIdx.x;     // 0..31
  const int half = lane >> 4;       // K-half selector for A/B operands
  const int lid  = lane & 15;
  const int mt = blockIdx.y;
  const int nt = blockIdx.x;

  // A (MxK): lane<16 holds row M=lid, K = 4k + {0,1}; lane>=16: K = 4k + {2,3}
  const float* ga = g + (size_t)(mt * 16 + lid) * KDIM + (half << 1);
  // B (KxN) mirrored layout; W^T is K-contiguous per column j
  const float* wb = w + (size_t)(nt * 16 + lid) * KDIM + (half << 1);

  v8f acc = {0.f, 0.f, 0.f, 0.f, 0.f, 0.f, 0.f, 0.f};
#pragma unroll 8
  for (int k = 0; k < KDIM; k += 4) {
    v2f a = *(const v2f*)(ga + k);
    v2f b = *(const v2f*)(wb + k);
    acc = __builtin_amdgcn_wmma_f32_16x16x4_f32(
        /*neg_a=*/false, a, /*neg_b=*/false, b,
        /*c_mod=*/(short)0, acc, /*reuse_a=*/false, /*reuse_b=*/false);
  }

  const float st = fminf(fmaxf(strd[0], 0.0f), (float)N_);
  const int col = nt * 16 + lid;               // j (re/im interleaved)
  const bool isReal = (col & 1) == 0;

#pragma unroll
  for (int r = 0; r < 8; ++r) {
    float v  = acc[r];
    float pv = __shfl_xor(v, 1, 32);           // partner re<->im (adjacent lane)
    int row = mt * 16 + r + (half << 3);       // C/D layout: +8 rows in upper half
    if (row < M_ && col < JN) {
      int f = col >> 1;
      int b = row / T_;
      int t = row - b * T_;
      float frame = st * (float)t;
      float frac = frame - floorf(frame);
      float ang = (TWO_PI / (float)N_) * frac * (float)f;   // shift[t,f]
      float c = cosf(ang), s = sinf(ang);
      float zr = isReal ? v : pv;
      float zi = isReal ? pv : v;
      float outr = c * zr - s * zi;
      float outi = s * zr + c * zi;
      size_t o = ((size_t)b * F_ + f) * T_ + t;
      if (isReal) {
        out[o] = sqrtf(outr * outr + outi * outi) + EPSF;   // spec
        out[(size_t)B_ * F_ * T_ + 2 * o] = outr;           // stft.re
      } else {
        out[(size_t)B_ * F_ * T_ + 2 * o + 1] = outi;       // stft.im
      }
    }
  }
}

extern "C" void kernel_launch(void* const* d_in, const int* in_sizes, int n_in,
                              void* d_out, int out_size, void* d_ws, size_t ws_size,
                              hipStream_t stream) {
  (void)in_sizes; (void)n_in; (void)out_size; (void)ws_size;
  const float* x    = (const float*)d_in[0];
  const float* win  = (const float*)d_in[1];
  const float* strd = (const float*)d_in[2];
  float* out = (float*)d_out;

  float* g = (float*)d_ws;                       // MPAD*KDIM floats (2.56 MB)
  float* w = g + (size_t)MPAD * KDIM;            // JPAD*KDIM floats (1.08 MB)

  int ge = MPAD * KDIM;
  build_g<<<(ge + 255) / 256, 256, 0, stream>>>(x, win, strd, g);
  int we = JPAD * KDIM;
  build_w<<<(we + 255) / 256, 256, 0, stream>>>(w);

  dim3 grid(NT, MT);                              // 33 x 78 tiles, 1 wave each
  wmma_dft<<<grid, 32, 0, stream>>>(g, w, strd, out);
}